// gcnBlock_22720376996373
// MI455X (gfx1250) — compile-verified
//
#include <hip/hip_runtime.h>
#include <hip/hip_bf16.h>

typedef __attribute__((ext_vector_type(2))) float v2f;
typedef __attribute__((ext_vector_type(8))) float v8f;

#define LN_EPS 1e-5f

// ---------------- degree kernels ----------------
__global__ void initDegKernel(float* __restrict__ deg, int N) {
    int i = blockIdx.x * blockDim.x + threadIdx.x;
    if (i < N) deg[i] = 1.0f;  // self loop
}

__global__ void accumDegKernel(const long long* __restrict__ ei, float* __restrict__ deg,
                               int E) {
    int e = blockIdx.x * blockDim.x + threadIdx.x;
    if (e < E) {
        int d = (int)ei[(size_t)E + e];  // dst row
        unsafeAtomicAdd(&deg[d], 1.0f);
    }
}

__global__ void finishDinvKernel(float* __restrict__ deg, int N) {
    int i = blockIdx.x * blockDim.x + threadIdx.x;
    if (i < N) deg[i] = rsqrtf(deg[i]);  // deg >= 1 always
}

// ---------------- WMMA fp32 GEMM: h = x @ W, D_IN = D_OUT = 64 ----------------
// One wave computes a 16-row x 64-col strip of h; 8 waves per 256-thread block.
// W is staged into LDS pre-swizzled so that each B fragment (4x16 f32 tile slice)
// is a contiguous 8-byte float2 per lane:
//   Ws2[((kstep*2 + hi)*4 + c)*16 + m] = { W[kstep*4+2*hi][c*16+m],
//                                          W[kstep*4+2*hi+1][c*16+m] }
// A frag (16x4 f32): lane m holds row r0+m; VGPR0 = K(2*hi), VGPR1 = K(2*hi+1)
// D (16x16 f32): VGPR v holds row v + 8*hi, col = lane&15
__global__ __launch_bounds__(256) void gemmWmmaKernel(const float* __restrict__ x,
                                                      const float* __restrict__ W,
                                                      float* __restrict__ h, int N) {
    __shared__ v2f Ws2[2048];                       // 16 KB, swizzled W image
    {
        float* Wf = (float*)Ws2;
        for (int idx = threadIdx.x; idx < 4096; idx += 256) {
            int k = idx >> 6, n = idx & 63;
            int kstep = k >> 2, khi = (k >> 1) & 1, j = k & 1;
            int c = n >> 4, m = n & 15;
            Wf[(((kstep * 2 + khi) * 4 + c) * 16 + m) * 2 + j] = W[idx];
        }
    }
    __syncthreads();

    const int wave = threadIdx.x >> 5;
    const int lane = threadIdx.x & 31;
    const int strip = blockIdx.x * 8 + wave;        // 16-row strip index
    if (strip * 16 >= N) return;                    // wave-uniform: EXEC all-1 inside

    const int r0 = strip * 16;
    const int m  = lane & 15;
    const int hi = lane >> 4;                       // 0: lanes 0-15, 1: lanes 16-31

    v8f acc0 = {}, acc1 = {}, acc2 = {}, acc3 = {};
    const float* xrow = x + (size_t)(r0 + m) * 64;

    #pragma unroll
    for (int kstep = 0; kstep < 16; ++kstep) {
        v2f a = *(const v2f*)(xrow + kstep * 4 + 2 * hi);   // (K, K+1), 8B aligned

        const v2f* wb = Ws2 + (size_t)((kstep * 2 + hi) * 4) * 16 + m;
        v2f b0 = wb[0];     // tile c=0, one ds_load_b64 each, even-pair aligned
        v2f b1 = wb[16];
        v2f b2 = wb[32];
        v2f b3 = wb[48];

        acc0 = __builtin_amdgcn_wmma_f32_16x16x4_f32(false, a, false, b0, (short)0, acc0, false, false);
        acc1 = __builtin_amdgcn_wmma_f32_16x16x4_f32(false, a, false, b1, (short)0, acc1, false, false);
        acc2 = __builtin_amdgcn_wmma_f32_16x16x4_f32(false, a, false, b2, (short)0, acc2, false, false);
        acc3 = __builtin_amdgcn_wmma_f32_16x16x4_f32(false, a, false, b3, (short)0, acc3, false, false);
    }

    float* hout = h + (size_t)r0 * 64;
    #pragma unroll
    for (int v = 0; v < 8; ++v) {
        const int row = v + 8 * hi;
        hout[row * 64 + m]      = acc0[v];
        hout[row * 64 + 16 + m] = acc1[v];
        hout[row * 64 + 32 + m] = acc2[v];
        hout[row * 64 + 48 + m] = acc3[v];
    }
}

// ---------------- self-loop init: agg[n][f] = h[n][f] * dinv[n]^2 ----------------
__global__ void selfLoopKernel(const float* __restrict__ h, const float* __restrict__ dinv,
                               float* __restrict__ agg, int N) {
    long long t = (long long)blockIdx.x * blockDim.x + threadIdx.x;   // N*32 threads
    if (t < (long long)N * 32) {
        int n = (int)(t >> 5);
        float di = dinv[n];
        float c = di * di;
        v2f hv = *(const v2f*)(h + t * 2);
        v2f r = {hv.x * c, hv.y * c};
        *(v2f*)(agg + t * 2) = r;
    }
}

// ---------------- edge scatter: agg[dst][f] += h[src][f] * dinv[src]*dinv[dst] ----
// 32 threads per edge, 2 features per lane (coalesced b64 gather of h[src]).
__global__ void scatterKernel(const float* __restrict__ h, const float* __restrict__ dinv,
                              const long long* __restrict__ ei, float* __restrict__ agg,
                              int E) {
    long long t = (long long)blockIdx.x * blockDim.x + threadIdx.x;
    int e  = (int)(t >> 5);
    int f2 = (int)(t & 31);                  // features 2*f2, 2*f2+1
    if (e < E) {
        int s = (int)ei[e];
        int d = (int)ei[(size_t)E + e];
        float c = dinv[s] * dinv[d];
        v2f hv = *(const v2f*)(h + (size_t)s * 64 + 2 * f2);
        float* dst = agg + (size_t)d * 64 + 2 * f2;
        unsafeAtomicAdd(dst,     hv.x * c);
        unsafeAtomicAdd(dst + 1, hv.y * c);
    }
}

// ---------------- LayerNorm + ReLU, one node per wave32, in place on d_out ----------------
__global__ __launch_bounds__(256) void lnReluKernel(float* __restrict__ agg,
                                                    const float* __restrict__ b,
                                                    const float* __restrict__ gamma,
                                                    const float* __restrict__ beta,
                                                    int N) {
    const int wave = threadIdx.x >> 5;
    const int lane = threadIdx.x & 31;
    const int node = blockIdx.x * 8 + wave;
    if (node >= N) return;

    float* row = agg + (size_t)node * 64;
    float v0 = row[lane]      + b[lane];
    float v1 = row[lane + 32] + b[lane + 32];

    float s1 = v0 + v1;
    float s2 = v0 * v0 + v1 * v1;
    #pragma unroll
    for (int mm = 16; mm >= 1; mm >>= 1) {
        s1 += __shfl_xor(s1, mm, 32);
        s2 += __shfl_xor(s2, mm, 32);
    }
    float mu  = s1 * (1.0f / 64.0f);
    float var = s2 * (1.0f / 64.0f) - mu * mu;
    float inv = rsqrtf(var + LN_EPS);

    float r0 = (v0 - mu) * inv * gamma[lane]      + beta[lane];
    float r1 = (v1 - mu) * inv * gamma[lane + 32] + beta[lane + 32];
    row[lane]      = fmaxf(r0, 0.0f);
    row[lane + 32] = fmaxf(r1, 0.0f);
}

extern "C" void kernel_launch(void* const* d_in, const int* in_sizes, int n_in,
                              void* d_out, int out_size, void* d_ws, size_t ws_size,
                              hipStream_t stream) {
    const float*     x     = (const float*)d_in[0];
    const float*     W     = (const float*)d_in[1];
    const float*     b     = (const float*)d_in[2];
    const float*     gamma = (const float*)d_in[3];
    const float*     beta  = (const float*)d_in[4];
    const long long* ei    = (const long long*)d_in[5];

    const int N = in_sizes[0] / 64;   // 50000
    const int E = in_sizes[5] / 2;    // 800000

    // workspace layout: [deg/dinv: N][pad to 50176][h: N*64]
    float* deg = (float*)d_ws;
    float* h   = (float*)d_ws + 50176;
    float* agg = (float*)d_out;       // accumulate directly into output buffer

    const int T = 256;

    initDegKernel<<<(N + T - 1) / T, T, 0, stream>>>(deg, N);
    accumDegKernel<<<(E + T - 1) / T, T, 0, stream>>>(ei, deg, E);
    finishDinvKernel<<<(N + T - 1) / T, T, 0, stream>>>(deg, N);

    // 8 strips of 16 rows per block -> 128 rows/block
    const int strips = (N + 15) / 16;                // 3125
    gemmWmmaKernel<<<(strips + 7) / 8, T, 0, stream>>>(x, W, h, N);

    long long selfWork = (long long)N * 32;
    selfLoopKernel<<<(int)((selfWork + T - 1) / T), T, 0, stream>>>(h, deg, agg, N);

    long long edgeWork = (long long)E * 32;
    scatterKernel<<<(int)((edgeWork + T - 1) / T), T, 0, stream>>>(h, deg, ei, agg, E);

    lnReluKernel<<<(N + 7) / 8, T, 0, stream>>>(agg, b, gamma, beta, N);
}